// LRPResNet34_39427799777441
// MI455X (gfx1250) — compile-verified
//
#include <hip/hip_runtime.h>

// LRP-epsilon through Conv2d 3x3/s1/p1, NCHW, fp32:
//   pass 0: reorder w into per-tap, pair-interleaved [tap][K/2][M] float2 layouts
//           (one for the forward GEMM, one channel-transposed for the input-VJP)
//   pass 1: z = conv(x,w); s = R/(z+eps*sign z)        -> d_ws
//   pass 2: c = conv-input-VJP(s,w); out = x*c         -> d_out
// GEMM core: V_WMMA_F32_16X16X4_F32 (full fp32), K = 9 taps x 128 ch.
// Per-tap weights double-buffered in LDS via async global->LDS b64 (ASYNCcnt),
// so the weight DMA for tap t+1 runs shadowed under tap t's 224 WMMAs/wave.

typedef float v2f __attribute__((ext_vector_type(2)));
typedef float v8f __attribute__((ext_vector_type(8)));

#define LRP_EPS 1e-6f
#define C_CH    128
#define PAIRS   64           // C_CH/2 channel pairs (K-pairs)
#define HW      56
#define NTILES  7            // 7*16 = 112 = 2 rows * 56 cols per block
#define MPAD    144          // ws2 M-stride in v2f  (2*144 mod 64 == 32 -> conflict-free)
#define COLPAD  60           // xs2 col-stride in v2f (8*60 mod 64 == 32 -> conflict-free)
#define XS2_CI  (4 * COLPAD) // xs2 pair-stride (4 halo rows)
#define WS2_ELEMS (PAIRS * MPAD)     // 9216  v2f per weight buffer
#define XS2_ELEMS (PAIRS * XS2_CI)   // 15360 v2f
#define LDS_BYTES ((2 * WS2_ELEMS + XS2_ELEMS) * sizeof(v2f))  // 270336 B of 320 KB WGP LDS
#define TAP_ELEMS (PAIRS * C_CH)     // 8192 v2f per tap per direction

// -------- pass 0: one-time weight reorder (w is 128x128x3x3) --------
__global__ __launch_bounds__(256)
void reorder_weights(const float* __restrict__ w,
                     v2f* __restrict__ wt_fwd,   // [9][64][128]: {w[co][2p][tap], w[co][2p+1][tap]}
                     v2f* __restrict__ wt_bwd)   // [9][64][128]: {w[2p][ci][tap], w[2p+1][ci][tap]}
{
    int idx = blockIdx.x * 256 + threadIdx.x;    // 9*64*128 = 73728 total
    if (idx >= 9 * TAP_ELEMS) return;
    int tap  = idx / TAP_ELEMS;
    int rem  = idx - tap * TAP_ELEMS;
    int p    = rem >> 7;      // K pair
    int mcol = rem & 127;     // M index (co for fwd, ci for bwd)
    v2f f, g;
    f.x = w[((size_t)mcol * C_CH + 2 * p    ) * 9 + tap];   // fwd: K=ci, M=co
    f.y = w[((size_t)mcol * C_CH + 2 * p + 1) * 9 + tap];
    g.x = w[((size_t)(2 * p    ) * C_CH + mcol) * 9 + tap]; // bwd: K=co, M=ci
    g.y = w[((size_t)(2 * p + 1) * C_CH + mcol) * 9 + tap];
    wt_fwd[idx] = f;
    wt_bwd[idx] = g;
}

// -------- main implicit-GEMM kernel (used for both passes) --------
template <bool BACKWARD>
__global__ __launch_bounds__(256)
void lrp_conv_gemm(const v2f*  __restrict__ wt,   // reordered per-tap weights [9][64][128]
                   const float* __restrict__ src, // B-operand image: x (fwd) / s (bwd)
                   const float* __restrict__ aux, // relevance (fwd) / x (bwd)
                   float* __restrict__ dst)       // s (fwd) / out (bwd)
{
    extern __shared__ v2f smem[];
    v2f* wbuf0 = smem;                   // weight buffer A: [K-pair][M] stride MPAD
    v2f* wbuf1 = smem + WS2_ELEMS;       // weight buffer B (double buffer)
    v2f* xs2   = smem + 2 * WS2_ELEMS;   // input strip: [ci-pair][4 rows][COLPAD]

    const int strip = blockIdx.x;       // 0..27 -> output rows [2*strip, 2*strip+1]
    const int b     = blockIdx.y;       // 0..31
    const int h0    = strip * 2;
    const int tid   = threadIdx.x;
    const int lane  = tid & 31;
    const int wave  = tid >> 5;
    const int hs    = lane >> 4;        // half-wave: selects K pair (ISA 7.12.2 A/B layout)
    const int l15   = lane & 15;

    // async global->LDS b64 staging of one tap's 64x128 v2f weight slice
    auto stage_tap = [&](int tap, v2f* dstbuf) {
        const v2f* gbase = wt + (size_t)tap * TAP_ELEMS;
        for (int e = tid; e < TAP_ELEMS; e += 256) {
            int p    = e >> 7;
            int mcol = e & 127;
            unsigned ldsoff = (unsigned)(size_t)(const void*)(dstbuf + p * MPAD + mcol);
            const v2f* g = gbase + e;
            asm volatile("global_load_async_to_lds_b64 %0, %1, off"
                         :: "v"(ldsoff), "v"(g) : "memory");
        }
    };

    // ---- kick off tap 0 weight DMA; it overlaps the input-strip staging below ----
    stage_tap(0, wbuf0);

    // ---- stage input strip (rows h0-1..h0+2, cols -1..56, channel pairs) into LDS ----
    {
        const float* base = src + (size_t)b * C_CH * HW * HW;
        const int total = PAIRS * 4 * 58;
        for (int e = tid; e < total; e += 256) {
            int p   = e / (4 * 58);
            int rem = e - p * (4 * 58);
            int row = rem / 58;
            int col = rem - row * 58;
            int h   = h0 - 1 + row;
            int gw  = col - 1;
            v2f v = {0.0f, 0.0f};
            if ((unsigned)h < HW && (unsigned)gw < HW) {
                size_t gi = ((size_t)(2 * p) * HW + h) * HW + gw;
                v.x = base[gi];
                v.y = base[gi + HW * HW];
            }
            xs2[p * XS2_CI + row * COLPAD + col] = v;
        }
    }
    asm volatile("s_wait_asynccnt 0x0" ::: "memory");
    __syncthreads();

    // per-lane N coordinates (row, col within 2x56 strip) for each tile
    int rarr[NTILES], carr[NTILES];
#pragma unroll
    for (int t = 0; t < NTILES; ++t) {
        int n = t * 16 + l15;
        rarr[t] = n / HW;
        carr[t] = n - rarr[t] * HW;
    }

    v8f acc[NTILES] = {};
    const int m = wave * 16 + l15;      // A-fragment M index for this lane

    for (int tap = 0; tap < 9; ++tap) {
        v2f* cur = (tap & 1) ? wbuf1 : wbuf0;
        v2f* nxt = (tap & 1) ? wbuf0 : wbuf1;

        // issue next tap's weight DMA now; it runs shadowed under this tap's WMMAs.
        // (nxt was last read at tap-1; the barrier at the end of tap-1 ordered it.)
        if (tap < 8) {
            stage_tap(tap + 1, nxt);
            if (tap < 7)   // warm L2 for the tap after next (global_prefetch_b8)
                __builtin_prefetch(wt + (size_t)(tap + 2) * TAP_ELEMS, 0, 0);
        }

        const int kh = tap / 3, kw = tap - 3 * (tap / 3);
        // halo offsets: fwd shift (kh-1,kw-1) -> +kh; bwd flipped -> +2-kh
        const int roff = BACKWARD ? (2 - kh) : kh;
        const int coff = BACKWARD ? (2 - kw) : kw;

        for (int k = 0; k < 32; ++k) {
            const int kp = k * 2 + hs;                 // K pair: covers K = 4k+2hs, +1
            v2f a = cur[kp * MPAD + m];                // one ds_load_b64
#pragma unroll
            for (int t = 0; t < NTILES; ++t) {
                v2f bb = xs2[kp * XS2_CI + (rarr[t] + roff) * COLPAD + (carr[t] + coff)];
                acc[t] = __builtin_amdgcn_wmma_f32_16x16x4_f32(
                    false, a, false, bb, (short)0, acc[t], false, false);
            }
        }

        // own async DMA done, then block-wide ordering before buffer swap/reuse
        asm volatile("s_wait_asynccnt 0x0" ::: "memory");
        __syncthreads();
    }

    // ---- epilogue: C layout (ISA 7.12.2): VGPR r -> M = r (lanes 0-15) / r+8 ----
    const size_t chstride = (size_t)HW * HW;
#pragma unroll
    for (int t = 0; t < NTILES; ++t) {
        const int h  = h0 + rarr[t];
        const int ww = carr[t];
#pragma unroll
        for (int r = 0; r < 8; ++r) {
            const int mm = wave * 16 + hs * 8 + r;
            const size_t idx = ((size_t)(b * C_CH + mm)) * chstride + (size_t)h * HW + ww;
            float v = acc[t][r];
            if (BACKWARD) {
                dst[idx] = aux[idx] * v;                         // out = x * c
            } else {
                float zst = v + (v >= 0.0f ? LRP_EPS : -LRP_EPS);
                dst[idx] = aux[idx] / zst;                       // s = R/(z+eps*sign z)
            }
        }
    }
}

extern "C" void kernel_launch(void* const* d_in, const int* in_sizes, int n_in,
                              void* d_out, int out_size, void* d_ws, size_t ws_size,
                              hipStream_t stream) {
    const float* x   = (const float*)d_in[0];  // [32,128,56,56]
    const float* w   = (const float*)d_in[1];  // [128,128,3,3]
    const float* rel = (const float*)d_in[2];  // [32,128,56,56]
    float* out = (float*)d_out;

    (void)in_sizes; (void)n_in; (void)out_size; (void)ws_size;

    // workspace layout: s buffer (51.4 MB) | wt_fwd (590 KB) | wt_bwd (590 KB)
    const size_t S_ELEMS = 32ull * C_CH * HW * HW;     // 12,845,056 floats
    float* sbuf  = (float*)d_ws;
    v2f* wt_fwd  = (v2f*)(sbuf + S_ELEMS);
    v2f* wt_bwd  = wt_fwd + 9 * TAP_ELEMS;

    hipFuncSetAttribute(reinterpret_cast<const void*>(&lrp_conv_gemm<false>),
                        hipFuncAttributeMaxDynamicSharedMemorySize, (int)LDS_BYTES);
    hipFuncSetAttribute(reinterpret_cast<const void*>(&lrp_conv_gemm<true>),
                        hipFuncAttributeMaxDynamicSharedMemorySize, (int)LDS_BYTES);

    // pass 0: reorder weights into per-tap GEMM layouts
    reorder_weights<<<dim3((9 * TAP_ELEMS + 255) / 256), dim3(256), 0, stream>>>(
        w, wt_fwd, wt_bwd);

    dim3 grid(28, 32);   // 28 row-strips x 32 batches
    dim3 block(256);     // 8 wave32

    // pass 1: z = conv(x,w); s = R / (z + eps*sign z)  -> sbuf
    lrp_conv_gemm<false><<<grid, block, LDS_BYTES, stream>>>(wt_fwd, x, rel, sbuf);
    // pass 2: c = conv-input-VJP(s,w); out = x * c     -> d_out
    lrp_conv_gemm<true ><<<grid, block, LDS_BYTES, stream>>>(wt_bwd, sbuf, x, out);
}